// SpatialEmbeddingLoss_420906795566
// MI455X (gfx1250) — compile-verified
//
#include <hip/hip_runtime.h>
#include <cstddef>

typedef __attribute__((ext_vector_type(2))) float v2f;
typedef __attribute__((ext_vector_type(8))) float v8f;

static constexpr int Bn = 4;
static constexpr int Nn = 250000;
static constexpr int Kn = 32;
static constexpr int NTILES = Nn / 16;            // 15625
static constexpr int NBINS = 65536;
static constexpr size_t HKN = (size_t)Kn * NBINS; // 2,097,152 bins per array

// ---- fast device math (native v_exp_f32 / v_rcp_f32) ----
__device__ __forceinline__ float fast_exp(float x) { return __expf(x); }
__device__ __forceinline__ float fast_rcp(float x) { return __builtin_amdgcn_rcpf(x); }
// tanh(x) = 1 - 2/(exp(2x)+1); exp->inf => rcp->0 => +1, exp->0 => -1 (overflow-safe)
__device__ __forceinline__ float fast_tanh(float x) {
    return 1.f - 2.f * fast_rcp(fast_exp(2.f * x) + 1.f);
}
__device__ __forceinline__ float fast_sigmoid(float x) {
    return fast_rcp(1.f + fast_exp(-x));
}

// ---------------- generic zero ----------------
__global__ void sel_zero_kernel(float* p, size_t n) {
    for (size_t i = blockIdx.x * (size_t)blockDim.x + threadIdx.x; i < n;
         i += (size_t)gridDim.x * blockDim.x)
        p[i] = 0.f;
}

// ---------------- stage 1: per-instance sums + seed_bg ----------------
// stats[(b*K+k)*8 + {0:count,1..3:sum emb,4..6:sum sigma,7:sum |sigma|^2}]
__global__ void sel_stats_kernel(const float* __restrict__ offsets,
                                 const float* __restrict__ coords,
                                 const float* __restrict__ sigmas,
                                 const float* __restrict__ seeds,
                                 const int* __restrict__ instances,
                                 float* __restrict__ stats,
                                 float* __restrict__ seedbg) {
    __shared__ float slot[Kn * 8];
    __shared__ float sbg;
    const int tid = threadIdx.x;
    const int b = blockIdx.y;
    if (tid < Kn * 8) slot[tid] = 0.f;
    if (tid == 0) sbg = 0.f;
    __syncthreads();
    float bgacc = 0.f;
    for (int n = blockIdx.x * blockDim.x + tid; n < Nn; n += gridDim.x * blockDim.x) {
        const size_t i = (size_t)b * Nn + n;
        const size_t i3 = i * 3;
        const int inst = instances[i];
        if (inst >= 0) {
            float s0 = sigmas[i3 + 0], s1 = sigmas[i3 + 1], s2 = sigmas[i3 + 2];
            float e0 = fast_tanh(offsets[i3 + 0]) + coords[i3 + 0];
            float e1 = fast_tanh(offsets[i3 + 1]) + coords[i3 + 1];
            float e2 = fast_tanh(offsets[i3 + 2]) + coords[i3 + 2];
            float* sl = slot + inst * 8;
            atomicAdd(sl + 0, 1.f);
            atomicAdd(sl + 1, e0); atomicAdd(sl + 2, e1); atomicAdd(sl + 3, e2);
            atomicAdd(sl + 4, s0); atomicAdd(sl + 5, s1); atomicAdd(sl + 6, s2);
            atomicAdd(sl + 7, s0 * s0 + s1 * s1 + s2 * s2);
        } else {
            float sd = fast_sigmoid(seeds[i]);
            bgacc += sd * sd;
        }
    }
    atomicAdd(&sbg, bgacc);
    __syncthreads();
    if (tid < Kn * 8) atomicAdd(&stats[(size_t)b * Kn * 8 + tid], slot[tid]);
    if (tid == 0) atomicAdd(&seedbg[b], sbg);
}

// ---------------- stage 2: per-(b,k) coefficients ----------------
// acoef row j: [se0,se1,se2, -2se*c (x3), sum se*c^2, 0]
// (precise expf here: s_e accuracy propagates into every quad)
__global__ void sel_coef_kernel(const float* __restrict__ stats,
                                float* __restrict__ acoef,
                                float* __restrict__ smoothv,
                                float* __restrict__ validf,
                                float* __restrict__ objv) {
    __shared__ float sobj[Bn];
    const int t = threadIdx.x; // 0..127
    if (t < Bn) sobj[t] = 0.f;
    __syncthreads();
    if (t < Bn * Kn) {
        const int b = t / Kn;
        const float* st = stats + (size_t)t * 8;
        const float count = st[0];
        const float vf = (count > 0.f) ? 1.f : 0.f;
        const float cnt = fmaxf(count, 1.f);
        const float c0 = st[1] / cnt, c1 = st[2] / cnt, c2 = st[3] / cnt;
        const float sk0 = st[4] / cnt, sk1 = st[5] / cnt, sk2 = st[6] / cnt;
        const float dev2 = st[7] - 2.f * (st[4] * sk0 + st[5] * sk1 + st[6] * sk2)
                         + count * (sk0 * sk0 + sk1 * sk1 + sk2 * sk2);
        smoothv[t] = dev2 / (cnt * 3.f);
        const float se0 = expf(10.f * sk0), se1 = expf(10.f * sk1), se2 = expf(10.f * sk2);
        float* a = acoef + (size_t)t * 8;
        a[0] = se0; a[1] = se1; a[2] = se2;
        a[3] = -2.f * se0 * c0; a[4] = -2.f * se1 * c1; a[5] = -2.f * se2 * c2;
        a[6] = se0 * c0 * c0 + se1 * c1 * c1 + se2 * c2 * c2;
        a[7] = 0.f;
        validf[t] = vf;
        atomicAdd(&sobj[b], vf);
    }
    __syncthreads();
    if (t < Bn) objv[t] = fmaxf(sobj[t], 1.f);
}

// ---------------- stage 3: WMMA quad -> probs -> error histograms ----------------
// quad tile: D(32x16) = A(32x8) * F(8x16), F = [e0^2,e1^2,e2^2,e0,e1,e2,1,0] per point.
// Each lane's D column equals the lane's own point (col = lane%16), so labels /
// seed / histogram updates are lane-local.
__global__ void sel_quad_hist_kernel(const float* __restrict__ offsets,
                                     const float* __restrict__ coords,
                                     const float* __restrict__ seeds,
                                     const int* __restrict__ instances,
                                     const float* __restrict__ acoef,
                                     unsigned* __restrict__ histCnt,
                                     unsigned* __restrict__ histGt,
                                     float* __restrict__ histE,
                                     float* __restrict__ seedfg,
                                     int b) {
    __shared__ float sfg;
    const int tid = threadIdx.x;
    if (tid == 0) sfg = 0.f;
    __syncthreads();
    const int lane = tid & 31;
    const int row = lane & 15;       // A-row / point column
    const int hh  = lane >> 4;       // lane half selects inner-K pair & D row offset
    const int waveGlobal = (blockIdx.x * blockDim.x + tid) >> 5;
    const int nWaves = (gridDim.x * blockDim.x) >> 5;

    // Load A operands once (constant across tiles). 16x4 f32 A layout:
    // lane: a.x = A[row][half?2:0], a.y = A[row][half?3:1] (+4 for chunk 1).
    const float* Abase = acoef + (size_t)b * Kn * 8;
    const float* r0 = Abase + row * 8;          // k-tile 0 (k = 0..15)
    const float* r1 = Abase + (row + 16) * 8;   // k-tile 1 (k = 16..31)
    const int j = hh * 2;
    v2f a00, a01, a10, a11;
    a00.x = r0[j];     a00.y = r0[j + 1];
    a01.x = r0[4 + j]; a01.y = r0[4 + j + 1];
    a10.x = r1[j];     a10.y = r1[j + 1];
    a11.x = r1[4 + j]; a11.y = r1[4 + j + 1];

    const v8f zc = {};
    float fgacc = 0.f;

    for (int t = waveGlobal; t < NTILES; t += nWaves) {
        const int n = t * 16 + row;
        const size_t i = (size_t)b * Nn + n;
        const size_t i3 = i * 3;
        __builtin_prefetch(&offsets[i3 + 48], 0, 1); // global_prefetch_b8
        const float e0 = fast_tanh(offsets[i3 + 0]) + coords[i3 + 0];
        const float e1 = fast_tanh(offsets[i3 + 1]) + coords[i3 + 1];
        const float e2 = fast_tanh(offsets[i3 + 2]) + coords[i3 + 2];

        // B operand (4x16 f32): lane: b.x = F[half?2:0][row], b.y = F[half?3:1][row]
        v2f b0, b1;
        if (hh) { b0.x = e2 * e2; b0.y = e0;      b1.x = 1.f; b1.y = 0.f; }
        else    { b0.x = e0 * e0; b0.y = e1 * e1; b1.x = e1;  b1.y = e2;  }

        v8f d0 = __builtin_amdgcn_wmma_f32_16x16x4_f32(false, a00, false, b0, (short)0, zc, false, false);
        d0     = __builtin_amdgcn_wmma_f32_16x16x4_f32(false, a01, false, b1, (short)0, d0, false, false);
        v8f d1 = __builtin_amdgcn_wmma_f32_16x16x4_f32(false, a10, false, b0, (short)0, zc, false, false);
        d1     = __builtin_amdgcn_wmma_f32_16x16x4_f32(false, a11, false, b1, (short)0, d1, false, false);

        const int inst = instances[i];
        const float sd = fast_sigmoid(seeds[i]);

#pragma unroll
        for (int v = 0; v < 8; ++v) {
#pragma unroll
            for (int kt = 0; kt < 2; ++kt) {
                const float q = (kt == 0) ? d0[v] : d1[v];
                const int k = kt * 16 + hh * 8 + v; // D row M = v + 8*half (+16 per k-tile)
                const float p = fast_exp(-q);       // native v_exp_f32
                const bool lab = (inst == k);
                const float e = lab ? (2.f - 2.f * p) : (2.f * p);
                int ba = (int)(e * (float)(NBINS / 2));
                ba = ba < 0 ? 0 : (ba > NBINS - 1 ? NBINS - 1 : ba);
                const int bin = (NBINS - 1) - ba;              // ascending bin == descending e
                const size_t h = (size_t)k * NBINS + bin;
                atomicAdd(&histCnt[h], 1u);
                atomicAdd(&histE[h], e);
                if (lab) {
                    atomicAdd(&histGt[h], 1u);
                    fgacc += (sd - p) * (sd - p);
                }
            }
        }
    }
    atomicAdd(&sfg, fgacc);
    __syncthreads();
    if (tid == 0) atomicAdd(&seedfg[b], sfg);
}

// ---------------- stage 4: binned Lovasz (jaccard monotone => tie-order free) ----------------
__global__ void sel_lovasz_kernel(const unsigned* __restrict__ histCnt,
                                  const unsigned* __restrict__ histGt,
                                  const float* __restrict__ histE,
                                  const float* __restrict__ stats,
                                  float* __restrict__ lov, int b) {
    const int k = blockIdx.x;
    const int t = threadIdx.x; // 256 threads
    constexpr int BPT = NBINS / 256;
    const unsigned* hc = histCnt + (size_t)k * NBINS;
    const unsigned* hg = histGt + (size_t)k * NBINS;
    const float* he = histE + (size_t)k * NBINS;
    __shared__ float sN[256], sG[256];
    __shared__ float sacc;
    float cn = 0.f, cg = 0.f;
    for (int i = 0; i < BPT; ++i) { cn += (float)hc[t * BPT + i]; cg += (float)hg[t * BPT + i]; }
    sN[t] = cn; sG[t] = cg;
    if (t == 0) sacc = 0.f;
    __syncthreads();
    if (t == 0) { // exclusive scan of per-thread totals
        float aN = 0.f, aG = 0.f;
        for (int i = 0; i < 256; ++i) {
            float tn = sN[i], tg = sG[i];
            sN[i] = aN; sG[i] = aG;
            aN += tn; aG += tg;
        }
    }
    __syncthreads();
    const float G = stats[((size_t)b * Kn + k) * 8 + 0];
    float cumN = sN[t], cumG = sG[t];
    auto Jf = [&](float n, float g) { return (n <= 0.f) ? 0.f : 1.f - (G - g) / (G + n - g); };
    float Jprev = Jf(cumN, cumG);
    float acc = 0.f;
    for (int i = 0; i < BPT; ++i) {
        const unsigned c = hc[t * BPT + i];
        if (c) {
            const float cf = (float)c;
            cumN += cf;
            cumG += (float)hg[t * BPT + i];
            const float J = Jf(cumN, cumG);
            acc += (he[t * BPT + i] / cf) * (J - Jprev); // relu(e)=e since e>=0
            Jprev = J;
        }
    }
    atomicAdd(&sacc, acc);
    __syncthreads();
    if (t == 0) lov[(size_t)b * Kn + k] = sacc;
}

// ---------------- stage 5: final reduction ----------------
__global__ void sel_final_kernel(const float* __restrict__ lov,
                                 const float* __restrict__ smoothv,
                                 const float* __restrict__ validf,
                                 const float* __restrict__ objv,
                                 const float* __restrict__ seedbg,
                                 const float* __restrict__ seedfg,
                                 float* __restrict__ out) {
    __shared__ float accb[Bn];
    __shared__ float tot;
    const int t = threadIdx.x;
    if (t < Bn) accb[t] = 0.f;
    if (t == 0) tot = 0.f;
    __syncthreads();
    if (t < Bn * Kn) {
        const int b = t / Kn;
        // W_INST = 1, W_SMOOTH = 10
        atomicAdd(&accb[b], (1.0f * lov[t] + 10.0f * smoothv[t]) * validf[t]);
    }
    __syncthreads();
    if (t < Bn) {
        const float inst_smooth = accb[t] / objv[t];
        const float seed_loss = (seedbg[t] + seedfg[t]) / (float)Nn; // W_FG=1 inside seedfg
        atomicAdd(&tot, inst_smooth + 10.f * seed_loss);             // W_SEED = 10
    }
    __syncthreads();
    if (t == 0) out[0] = tot / (float)Bn;
}

extern "C" void kernel_launch(void* const* d_in, const int* in_sizes, int n_in,
                              void* d_out, int out_size, void* d_ws, size_t ws_size,
                              hipStream_t stream) {
    (void)in_sizes; (void)n_in; (void)out_size; (void)ws_size;
    const float* offsets = (const float*)d_in[0];
    const float* coords = (const float*)d_in[1];
    const float* sigmas = (const float*)d_in[2];
    const float* seeds = (const float*)d_in[3];
    const int* instances = (const int*)d_in[4];
    float* out = (float*)d_out;

    // workspace layout: [histCnt u32 | histGt u32 | histE f32] (24 MB, reused per batch)
    //                   then small stats/coef/accumulator arrays
    unsigned* histCnt = (unsigned*)d_ws;
    unsigned* histGt = histCnt + HKN;
    float* histE = (float*)(histGt + HKN);
    float* stats = histE + HKN;            // B*K*8
    float* acoef = stats + Bn * Kn * 8;    // B*K*8
    float* smoothv = acoef + Bn * Kn * 8;  // B*K
    float* validf = smoothv + Bn * Kn;     // B*K
    float* lov = validf + Bn * Kn;         // B*K
    float* seedbg = lov + Bn * Kn;         // B
    float* seedfg = seedbg + Bn;           // B
    float* objv = seedfg + Bn;             // B
    const size_t smallFloats = (size_t)(Bn * Kn * 8) * 2 + (size_t)(Bn * Kn) * 3 + 3 * Bn;

    // zero atomically-accumulated small buffers (every call: deterministic)
    sel_zero_kernel<<<8, 256, 0, stream>>>(stats, smallFloats);

    // per-point segmented sums
    dim3 sg(128, Bn, 1);
    sel_stats_kernel<<<sg, 256, 0, stream>>>(offsets, coords, sigmas, seeds, instances,
                                             stats, seedbg);

    // per-(b,k) coefficients
    sel_coef_kernel<<<1, 128, 0, stream>>>(stats, acoef, smoothv, validf, objv);

    for (int b = 0; b < Bn; ++b) {
        sel_zero_kernel<<<2048, 256, 0, stream>>>((float*)histCnt, 3 * HKN);
        sel_quad_hist_kernel<<<240, 256, 0, stream>>>(offsets, coords, seeds, instances,
                                                      acoef, histCnt, histGt, histE,
                                                      seedfg, b);
        sel_lovasz_kernel<<<Kn, 256, 0, stream>>>(histCnt, histGt, histE, stats, lov, b);
    }

    sel_final_kernel<<<1, 128, 0, stream>>>(lov, smoothv, validf, objv, seedbg, seedfg, out);
}